// ConditionalPatchSlicedTransport_discrete_37151467110992
// MI455X (gfx1250) — compile-verified
//
#include <hip/hip_runtime.h>
#include <cstddef>

// ---------------------------------------------------------------------------
// ConditionalPatchSlicedTransport on MI455X (gfx1250)
//
// Pipeline (3 kernels on `stream`):
//   1) fwd:    data0 = blockwise data @ W      (V_WMMA_F32_16X16X4_F32 x4)
//   2) spline: delta = spline(data0) - data0, logj[n] = sum lj  (in-place in ws)
//   3) bwd:    out   = data + blockwise delta @ W^T  (WMMA again, fused add)
//
// Sizes: N=2048, NDIM=3072, NKERNEL=192 blocks of 16x16, NBIN=200, NCLASS=10.
// Memory-bound (~125MB HBM): fp32 throughout, WMMA f32 16x16x4 for the GEMMs.
// ---------------------------------------------------------------------------

typedef __attribute__((ext_vector_type(2))) float v2f;
typedef __attribute__((ext_vector_type(8))) float v8f;

#define NSAMP   2048
#define NDIMS   3072
#define NKER    192
#define NBINS   200

__device__ inline v8f wmma_f32_16x16x4(v2f a, v2f b, v8f c) {
#if defined(__HIP_DEVICE_COMPILE__)
  // 8 args: (neg_a, A, neg_b, B, c_mod, C, reuse_a, reuse_b)
  return __builtin_amdgcn_wmma_f32_16x16x4_f32(false, a, false, b, (short)0, c,
                                               false, false);
#else
  (void)a; (void)b;
  return c;
#endif
}

// Patch gather index: block k -> channel c, patch (nh, nw); element i = kh*4+kw
// dim index = (nh*4+kh)*96 + (nw*4+kw)*3 + c
__device__ inline int block_base(int k) {
  int c = k % 3;
  int p = k / 3;
  int nh = p >> 3;
  int nw = p & 7;
  return nh * 384 + nw * 12 + c;
}

// --------------------------------------------------------------------------
// Kernel 1: data0[n, k*16+j] = sum_i data[n, row(k,i)] * W[k][i][j]
// grid (24, 128), block 256 (8 waves; wave w handles block k = bx*8+w,
// sample tile m0 = by*16).
// --------------------------------------------------------------------------
__global__ void __launch_bounds__(256)
fwd_kernel(const float* __restrict__ data,
           const float* __restrict__ wblocks,
           float* __restrict__ data0) {
  const int lane = threadIdx.x & 31;
  const int wv   = threadIdx.x >> 5;
  const int k    = blockIdx.x * 8 + wv;           // 0..191
  const int m0   = blockIdx.y * 16;               // sample tile base
  const int half = lane >> 4;                     // 0: lanes 0-15, 1: 16-31
  const int mn   = lane & 15;

  const int base = block_base(k);
  const float* __restrict__ W = wblocks + k * 256;   // [i][j] row-major

  v8f acc = {};
#pragma unroll
  for (int kc = 0; kc < 4; ++kc) {
    const int i0 = 4 * kc + 2 * half;             // K-pair this half-wave owns
    // A (16x4 f32): M = sample (mn), K = patch-dim i0, i0+1
    const int kh = i0 >> 2, kw = i0 & 3;          // i0 even => kw in {0,2}
    const int ai = (m0 + mn) * NDIMS + base + kh * 96 + kw * 3;
    v2f a;
    a.x = data[ai];
    a.y = data[ai + 3];                           // kw+1 -> +3 floats
    // B (4x16 f32): K = i0/i0+1, N = comp j (mn)
    v2f b;
    b.x = W[i0 * 16 + mn];
    b.y = W[(i0 + 1) * 16 + mn];
    acc = wmma_f32_16x16x4(a, b, acc);
  }

  // D 16x16: VGPR r -> M = r + 8*half, N = mn. Coalesced store.
  const int gbase = (m0 + 8 * half) * NDIMS + k * 16 + mn;
#pragma unroll
  for (int r = 0; r < 8; ++r)
    data0[gbase + r * NDIMS] = acc[r];
}

// --------------------------------------------------------------------------
// Kernel 2: per-sample RQ spline + log-Jacobian.
// grid 2048, block 256. In-place: buf holds data0 in, delta = y - data0 out.
// --------------------------------------------------------------------------
__global__ void __launch_bounds__(256)
spline_kernel(float* __restrict__ buf,
              const int* __restrict__ label,
              const float* __restrict__ knots_x,
              const float* __restrict__ knots_y,
              const float* __restrict__ knots_d,
              float* __restrict__ logj) {
  const int n = blockIdx.x;
  const int c = label[n];
  const size_t crow = (size_t)c * NDIMS;

  float ljsum = 0.0f;
  for (int t = threadIdx.x; t < NDIMS; t += 256) {
    const float x = buf[(size_t)n * NDIMS + t];
    const float* __restrict__ xx = knots_x + (crow + t) * NBINS;
    const float* __restrict__ yy = knots_y + (crow + t) * NBINS;
    const float* __restrict__ dd = knots_d + (crow + t) * NBINS;
    if (t + 256 < NDIMS) {                  // global_prefetch_b8 next knot rows
      __builtin_prefetch(xx + 256 * NBINS, 0, 1);
      __builtin_prefetch(yy + 256 * NBINS, 0, 1);
      __builtin_prefetch(dd + 256 * NBINS, 0, 1);
    }

    const float lo = xx[0];
    const float hi = xx[NBINS - 1];
    float y, lj;
    if (x < lo) {
      const float d0 = dd[0];
      y  = yy[0] + (x - lo) * d0;
      lj = __logf(d0);
    } else if (x > hi) {
      const float dK = dd[NBINS - 1];
      y  = yy[NBINS - 1] + (x - hi) * dK;
      lj = __logf(dK);
    } else {
      // searchsorted(xx, x, 'right') - 1, clipped to [0, NBINS-2]
      int loi = 0, hii = NBINS;
      while (loi < hii) {
        const int mid = (loi + hii) >> 1;
        if (xx[mid] <= x) loi = mid + 1; else hii = mid;
      }
      int kk = loi - 1;
      kk = kk < 0 ? 0 : (kk > NBINS - 2 ? NBINS - 2 : kk);
      const float xk = xx[kk],  xk1 = xx[kk + 1];
      const float yk = yy[kk],  yk1 = yy[kk + 1];
      const float dk = dd[kk],  dk1 = dd[kk + 1];
      const float wdt = xk1 - xk;
      const float s   = (yk1 - yk) / wdt;
      const float xi  = (x - xk) / wdt;
      const float om  = 1.0f - xi;
      const float den = s + (dk1 + dk - 2.0f * s) * xi * om;
      y  = yk + (yk1 - yk) * (s * xi * xi + dk * xi * om) / den;
      lj = 2.0f * __logf(s)
         + __logf(dk1 * xi * xi + 2.0f * s * xi * om + dk * om * om)
         - 2.0f * __logf(den);
    }
    buf[(size_t)n * NDIMS + t] = y - x;     // delta, in place
    ljsum += lj;
  }

  __shared__ float red[256];
  red[threadIdx.x] = ljsum;
  __syncthreads();
#pragma unroll
  for (int s = 128; s > 0; s >>= 1) {
    if (threadIdx.x < (unsigned)s) red[threadIdx.x] += red[threadIdx.x + s];
    __syncthreads();
  }
  if (threadIdx.x == 0) logj[n] = red[0];
}

// --------------------------------------------------------------------------
// Kernel 3: out[n, row(k,i)] = data[n, row(k,i)] + sum_j delta[n,k*16+j]*W[k][i][j]
// Same wave mapping as fwd; B = W^T so both A and B load as aligned float2.
// --------------------------------------------------------------------------
__global__ void __launch_bounds__(256)
bwd_kernel(const float* __restrict__ data,
           const float* __restrict__ wblocks,
           const float* __restrict__ delta,
           float* __restrict__ out) {
  const int lane = threadIdx.x & 31;
  const int wv   = threadIdx.x >> 5;
  const int k    = blockIdx.x * 8 + wv;
  const int m0   = blockIdx.y * 16;
  const int half = lane >> 4;
  const int mn   = lane & 15;

  const int base = block_base(k);
  const float* __restrict__ W = wblocks + k * 256;

  v8f acc = {};
#pragma unroll
  for (int kc = 0; kc < 4; ++kc) {
    const int j0 = 4 * kc + 2 * half;             // K = comp j
    // A (16x4): M = sample, K = j; contiguous -> float2
    v2f a = *(const v2f*)(delta + (m0 + mn) * NDIMS + k * 16 + j0);
    // B (4x16): K = j, N = patch-dim i (mn); B[j][i] = W[i][j] -> row of W
    v2f b = *(const v2f*)(W + mn * 16 + j0);
    acc = wmma_f32_16x16x4(a, b, acc);
  }

  const int col = base + (mn >> 2) * 96 + (mn & 3) * 3;   // row(k, i=mn)
#pragma unroll
  for (int r = 0; r < 8; ++r) {
    const int gi = (m0 + 8 * half + r) * NDIMS + col;
    out[gi] = data[gi] + acc[r];
  }
}

// --------------------------------------------------------------------------
extern "C" void kernel_launch(void* const* d_in, const int* in_sizes, int n_in,
                              void* d_out, int out_size, void* d_ws,
                              size_t ws_size, hipStream_t stream) {
  (void)in_sizes; (void)n_in; (void)out_size; (void)ws_size;
  const float* data    = (const float*)d_in[0];   // (2048, 3072) f32
  const int*   label   = (const int*)  d_in[1];   // (2048,) i32
  const float* wblocks = (const float*)d_in[2];   // (192, 16, 16) f32
  const float* kx      = (const float*)d_in[3];   // (10, 3072, 200) f32
  const float* ky      = (const float*)d_in[4];
  const float* kd      = (const float*)d_in[5];

  float* out  = (float*)d_out;                    // (2048, 3072) f32
  float* logj = out + (size_t)NSAMP * NDIMS;      // (2048,) f32
  float* buf  = (float*)d_ws;                     // 2048*3072 f32 = 25.2 MB

  dim3 gmm(NKER / 8, NSAMP / 16);                 // (24, 128)
  fwd_kernel<<<gmm, 256, 0, stream>>>(data, wblocks, buf);
  spline_kernel<<<NSAMP, 256, 0, stream>>>(buf, label, kx, ky, kd, logj);
  bwd_kernel<<<gmm, 256, 0, stream>>>(data, wblocks, buf, out);
}